// GraphConvNet_81793357185280
// MI455X (gfx1250) — compile-verified
//
#include <hip/hip_runtime.h>
#include <hip/hip_bf16.h>

typedef __attribute__((ext_vector_type(2))) float v2f;
typedef __attribute__((ext_vector_type(8))) float v8f;

#define B_    32
#define C_    32
#define V_    1024
#define L_    13
#define CIN   224         // (2*3+1)*32
#define COUT  64
#define NPB   (V_ * L_)   // 13312 elements per (b,c) plane

// gfx1250 async memory->LDS copy helpers (ASYNCcnt-tracked, no VGPR round-trip)
__device__ __forceinline__ void async_b128(unsigned ldsOff, const float* g) {
    asm volatile("global_load_async_to_lds_b128 %0, %1, off"
                 :: "v"(ldsOff), "v"(g) : "memory");
}
__device__ __forceinline__ void async_b32(unsigned ldsOff, const float* g) {
    asm volatile("global_load_async_to_lds_b32 %0, %1, off"
                 :: "v"(ldsOff), "v"(g) : "memory");
}

// ---------------------------------------------------------------------------
// nconv: Y[bc][w][l] = sum_v A[v][w] * X[bc][v][l]
// grid = (V_/128, B_*C_), block = 256 (8 wave32)
// K (v) staged in 32-deep chunks, double-buffered via async-to-LDS so the
// global fill of chunk k+1 overlaps the 64 WMMAs of chunk k.
// ---------------------------------------------------------------------------
__global__ __launch_bounds__(256) void nconv_kernel(const float* __restrict__ X,
                                                    const float* __restrict__ A,
                                                    float* __restrict__ Y) {
    const int wblk = blockIdx.x * 128;
    const int bc   = blockIdx.y;
    const float* Xp = X + (size_t)bc * NPB;
    float*       Yp = Y + (size_t)bc * NPB;

    __shared__ float ldsA[2][32 * 144];    // [v 32][w 128] pad->144
    __shared__ float ldsX[2][32 * 16];     // [v 32][l 16], cols 13..15 = 0

    const int tid  = threadIdx.x;
    const int lane = tid & 31;
    const int wave = tid >> 5;
    const int nl   = lane & 15;            // M row (A frag) / N col (B,C,D)
    const int half = lane >> 4;            // 0: K 0,1   1: K 2,3
    const int vsel = half * 2;
    const int wt   = wave * 16;

    // zero ONLY the pad columns (13..15) of both X buffers -> no overlap with
    // the async writes, visible after the first barrier
    for (int i = tid; i < 2 * 32 * 3; i += 256) {
        int b = i / 96, r = (i % 96) / 3, c = 13 + (i % 3);
        ldsX[b][r * 16 + c] = 0.0f;
    }

    // stage one 32-deep K chunk into buffer `buf`; exactly 6 async ops/thread
    auto stage = [&](int k0, int buf) {
#pragma unroll
        for (int i = 0; i < 4; ++i) {                 // A: 32 x 128 floats
            int f = tid + i * 256;                    // float4 slot 0..1023
            int row = f >> 5, c4 = f & 31;
            unsigned lo = (unsigned)(size_t)&ldsA[buf][row * 144 + c4 * 4];
            async_b128(lo, A + (size_t)(k0 + row) * V_ + wblk + c4 * 4);
        }
#pragma unroll
        for (int j = 0; j < 2; ++j) {                 // X: 32 x 13 floats
            int i = tid + j * 256; if (i > 415) i = 415;   // clamp: uniform cnt
            int row = i / 13, l = i - row * 13;
            unsigned lo = (unsigned)(size_t)&ldsX[buf][row * 16 + l];
            async_b32(lo, Xp + (size_t)(k0 + row) * L_ + l);
        }
    };

    stage(0, 0);
    v8f acc = {};

    for (int it = 0; it < 32; ++it) {
        const int buf = it & 1;
        if (it + 1 < 32) {
            stage((it + 1) * 32, buf ^ 1);            // prefetch next chunk
            asm volatile("s_wait_asynccnt 0x6" ::: "memory");  // chunk it done
        } else {
            asm volatile("s_wait_asynccnt 0x0" ::: "memory");
        }
        __syncthreads();                               // all waves' data landed

#pragma unroll
        for (int kk = 0; kk < 8; ++kk) {
            const int v0 = kk * 4 + vsel;
            v2f af, bf;
            af.x = ldsA[buf][(v0    ) * 144 + wt + nl];   // A^T tile (m=w,k=v)
            af.y = ldsA[buf][(v0 + 1) * 144 + wt + nl];
            bf.x = ldsX[buf][(v0    ) * 16 + nl];         // B tile (k=v,n=l)
            bf.y = ldsX[buf][(v0 + 1) * 16 + nl];
            acc = __builtin_amdgcn_wmma_f32_16x16x4_f32(
                false, af, false, bf, (short)0, acc, false, false);
        }
        __syncthreads();               // buffer drained before it is refilled
    }

    if (nl < L_) {
#pragma unroll
        for (int r = 0; r < 8; ++r) {
            const int m = r + half * 8;                   // C/D VGPR layout
            Yp[(size_t)(wblk + wt + m) * L_ + nl] = acc[r];
        }
    }
}

// ---------------------------------------------------------------------------
// pw: 1x1 conv block.  Out[b][o][n] (+)= sum_c W[o][c0+c] * H[b][c][n]
// grid = (NPB/64, B_), block = 256.  64(o) x 64(n) patch, K=32.
// ---------------------------------------------------------------------------
template <bool INIT>
__global__ __launch_bounds__(256) void pw_kernel(const float* __restrict__ H,
                                                 const float* __restrict__ W,
                                                 const float* __restrict__ bias,
                                                 int c0,
                                                 float* __restrict__ Out) {
    const int n0 = blockIdx.x * 64;
    const int b  = blockIdx.y;
    const float* Hp = H + (size_t)b * C_ * NPB;
    float*       Op = Out + (size_t)b * COUT * NPB;

    __shared__ float ldsW[64 * 36];   // [o 64][c 32] pad->36 (16B-aligned rows)
    __shared__ float ldsH[32 * 80];   // [c 32][n 64] pad->80

    const int tid  = threadIdx.x;
    const int lane = tid & 31;
    const int wave = tid >> 5;
    const int nl   = lane & 15;
    const int half = lane >> 4;
    const int vsel = half * 2;

#pragma unroll
    for (int j = 0; j < 2; ++j) {                 // W block: 512 float4
        int i = tid + j * 256;
        int o = i >> 3, c4 = i & 7;
        unsigned lo = (unsigned)(size_t)&ldsW[o * 36 + c4 * 4];
        async_b128(lo, W + o * CIN + c0 + c4 * 4);
    }
#pragma unroll
    for (int j = 0; j < 2; ++j) {                 // H patch: 512 float4
        int i = tid + j * 256;
        int c = i >> 4, n4 = i & 15;
        unsigned lo = (unsigned)(size_t)&ldsH[c * 80 + n4 * 4];
        async_b128(lo, Hp + (size_t)c * NPB + n0 + n4 * 4);
    }
    asm volatile("s_wait_asynccnt 0x0" ::: "memory");
    __syncthreads();

#pragma unroll
    for (int t = 0; t < 2; ++t) {
        const int tile = wave + t * 8;            // 16 tiles: 4(m) x 4(n)
        const int mt   = (tile & 3) * 16;
        const int nt   = (tile >> 2) * 16;
        v8f acc = {};
#pragma unroll
        for (int kk = 0; kk < 8; ++kk) {
            const int k = kk * 4 + vsel;
            v2f af, bf;
            af.x = ldsW[(mt + nl) * 36 + k];          // A tile (m=o, k=c)
            af.y = ldsW[(mt + nl) * 36 + k + 1];
            bf.x = ldsH[(k    ) * 80 + nt + nl];      // B tile (k=c, n)
            bf.y = ldsH[(k + 1) * 80 + nt + nl];
            acc = __builtin_amdgcn_wmma_f32_16x16x4_f32(
                false, af, false, bf, (short)0, acc, false, false);
        }
#pragma unroll
        for (int r = 0; r < 8; ++r) {
            const int o = mt + r + half * 8;
            const size_t idx = (size_t)o * NPB + n0 + nt + nl;
            if (INIT) Op[idx] = bias[o] + acc[r];
            else      Op[idx] += acc[r];
        }
    }
}

// ---------------------------------------------------------------------------
extern "C" void kernel_launch(void* const* d_in, const int* in_sizes, int n_in,
                              void* d_out, int out_size, void* d_ws, size_t ws_size,
                              hipStream_t stream) {
    const float* x    = (const float*)d_in[0];
    const float* A0   = (const float*)d_in[1];
    const float* A1   = (const float*)d_in[2];
    const float* A2   = (const float*)d_in[3];
    const float* W    = (const float*)d_in[4];
    const float* bias = (const float*)d_in[5];
    float* out = (float*)d_out;

    const size_t plane = (size_t)B_ * C_ * NPB;   // 13,631,488 floats
    float* x1 = (float*)d_ws;
    float* x2 = x1 + plane;

    const dim3 blk(256);
    const dim3 gA(V_ / 128, B_ * C_);             // 8 x 1024 WGs
    const dim3 gB(NPB / 64, B_);                  // 208 x 32 WGs

    // block 0: out = bias + W[:,0:32] @ x   (fully overwrites poisoned out)
    pw_kernel<true><<<gB, blk, 0, stream>>>(x, W, bias, 0, out);

    const float* As[3] = {A0, A1, A2};
    for (int k = 0; k < 3; ++k) {
        nconv_kernel<<<gA, blk, 0, stream>>>(x,  As[k], x1);   // x1 = A^T x
        nconv_kernel<<<gA, blk, 0, stream>>>(x1, As[k], x2);   // x2 = A^T x1
        pw_kernel<false><<<gB, blk, 0, stream>>>(x1, W, bias, 32 * (2 * k + 1), out);
        pw_kernel<false><<<gB, blk, 0, stream>>>(x2, W, bias, 32 * (2 * k + 2), out);
    }
}